// ShootingBlockMNModel1_85341000171551
// MI455X (gfx1250) — compile-verified
//
#include <hip/hip_runtime.h>

typedef __attribute__((ext_vector_type(2))) float v2f;
typedef __attribute__((ext_vector_type(8))) float v8f;

#define NSLOT 40          // 38 used + 2 pad
#define LDS_STRIDE 257    // 257 % 64 == 1 -> conflict-free rows/cols
#define LR_CONST 0.25f

// --- copy Theta_0 from inputs into ws ---------------------------------------
__global__ void init_theta_kernel(const float* __restrict__ t1,
                                  const float* __restrict__ b1,
                                  const float* __restrict__ t2,
                                  const float* __restrict__ b2,
                                  float* __restrict__ thetaAll) {
  if (threadIdx.x == 0) {
    thetaAll[0] = t1[0]; thetaAll[1] = t1[1]; thetaAll[2] = t1[2]; thetaAll[3] = t1[3];
    thetaAll[4] = b1[0]; thetaAll[5] = b1[1];
    thetaAll[6] = t2[0]; thetaAll[7] = t2[1]; thetaAll[8] = t2[2]; thetaAll[9] = t2[3];
    thetaAll[10] = b2[0]; thetaAll[11] = b2[1];
  }
}

// --- fused gradient + Hessian reduction pass at Theta_k ---------------------
// slots: 0-3 R1=p_a h_b | 4-7 R2=r_a q_b | 8-9 R3=r_a | 10-11 Sp=p_a
//        12-19 X1=p_a s_b q_e | 20-23 X2=p_a s_b | 24-31 X3=t_a q_b q_e
//        32-35 X4=t_a q_b | 36-37 X5=t_a
__global__ void pass_reduce_kernel(const float* __restrict__ inp, int K,
                                   const float* __restrict__ thetaAll, int kIdx,
                                   float* __restrict__ partials) {
  const float* th = thetaAll + kIdx * 12;
  const float T10 = th[0], T11 = th[1], T12 = th[2], T13 = th[3];
  const float T20 = th[6], T21 = th[7], T22 = th[8], T23 = th[9];
  const float b20 = th[10], b21 = th[11];

  float acc[NSLOT];
#pragma unroll
  for (int j = 0; j < NSLOT; ++j) acc[j] = 0.f;

  const float2* in2 = (const float2*)inp;
  const int stride = gridDim.x * blockDim.x;
  for (int i = blockIdx.x * blockDim.x + threadIdx.x; i < K; i += stride) {
    float2 qv = in2[i];
    float2 pv = in2[K + i];
    float q0 = qv.x, q1 = qv.y, p0 = pv.x, p1 = pv.y;
    float u0 = fmaf(T20, q0, fmaf(T21, q1, b20));
    float u1 = fmaf(T22, q0, fmaf(T23, q1, b21));
    float h0 = tanhf(u0), h1 = tanhf(u1);
    float s0 = 1.f - h0 * h0, s1 = 1.f - h1 * h1;
    float w0 = fmaf(T10, p0, T12 * p1);   // w = T1^T p
    float w1 = fmaf(T11, p0, T13 * p1);
    float r0 = s0 * w0, r1 = s1 * w1;
    float t0 = -2.f * h0 * r0, t1 = -2.f * h1 * r1;
    acc[0] += p0 * h0; acc[1] += p0 * h1; acc[2] += p1 * h0; acc[3] += p1 * h1;
    acc[4] += r0 * q0; acc[5] += r0 * q1; acc[6] += r1 * q0; acc[7] += r1 * q1;
    acc[8] += r0; acc[9] += r1; acc[10] += p0; acc[11] += p1;
    float ps00 = p0 * s0, ps01 = p0 * s1, ps10 = p1 * s0, ps11 = p1 * s1;
    acc[12] += ps00 * q0; acc[13] += ps00 * q1; acc[14] += ps01 * q0; acc[15] += ps01 * q1;
    acc[16] += ps10 * q0; acc[17] += ps10 * q1; acc[18] += ps11 * q0; acc[19] += ps11 * q1;
    acc[20] += ps00; acc[21] += ps01; acc[22] += ps10; acc[23] += ps11;
    float tq00 = t0 * q0, tq01 = t0 * q1, tq10 = t1 * q0, tq11 = t1 * q1;
    acc[24] += tq00 * q0; acc[25] += tq00 * q1; acc[26] += tq01 * q0; acc[27] += tq01 * q1;
    acc[28] += tq10 * q0; acc[29] += tq10 * q1; acc[30] += tq11 * q0; acc[31] += tq11 * q1;
    acc[32] += tq00; acc[33] += tq01; acc[34] += tq10; acc[35] += tq11;
    acc[36] += t0; acc[37] += t1;
  }

  // Stage per-lane partials to LDS as a 48 x 256 matrix (rows = slots).
  __shared__ float lds[48 * LDS_STRIDE];
  const int t = threadIdx.x;
#pragma unroll
  for (int j = 0; j < NSLOT; ++j) lds[j * LDS_STRIDE + t] = acc[j];
#pragma unroll
  for (int j = NSLOT; j < 48; ++j) lds[j * LDS_STRIDE + t] = 0.f;
  __syncthreads();

  // Wave 0: block reduction as matmul-by-ones using V_WMMA_F32_16X16X4_F32.
  // D = A(16x4) * ones(4x16) + C accumulated over 64 column chunks:
  // every column of D holds the 256-wide row sum.
  if (t < 32) {
    const int m = t & 15;           // A-matrix row for this lane
    const int kb = (t >> 4) * 2;    // K offset: lanes 0-15 -> K0,K1; 16-31 -> K2,K3
    v2f bo; bo[0] = 1.f; bo[1] = 1.f;   // all-ones B (layout-independent)
    for (int g = 0; g < 3; ++g) {
      v8f cacc = {};
      const int rowoff = (16 * g + m) * LDS_STRIDE + kb;
      for (int cc = 0; cc < 64; ++cc) {
        v2f av;
        av[0] = lds[rowoff + cc * 4];
        av[1] = lds[rowoff + cc * 4 + 1];
        cacc = __builtin_amdgcn_wmma_f32_16x16x4_f32(
            false, av, false, bo, (short)0, cacc, false, false);
      }
      // C/D layout: VGPR r, lanes 0-15 -> M=r (N=lane), lanes 16-31 -> M=r+8.
      if ((t & 15) == 0) {
        const int base = g * 16 + (t >> 4) * 8;
        float rs[8] = {cacc[0], cacc[1], cacc[2], cacc[3],
                       cacc[4], cacc[5], cacc[6], cacc[7]};
#pragma unroll
        for (int vr = 0; vr < 8; ++vr) {
          int slot = base + vr;
          if (slot < NSLOT) partials[blockIdx.x * NSLOT + slot] = rs[vr];
        }
      }
    }
  }
}

// --- combine block partials (fixed order) + GD update of Theta --------------
__global__ void reduce_update_kernel(const float* __restrict__ partials, int NB,
                                     float* __restrict__ sums, int kIdx,
                                     float* __restrict__ thetaAll,
                                     const float* __restrict__ invK,
                                     const float* __restrict__ invKb,
                                     float c) {
  __shared__ float s[NSLOT];
  const int j = threadIdx.x;
  if (j < NSLOT) {
    float tot = 0.f;
    for (int b = 0; b < NB; ++b) tot += partials[b * NSLOT + j];
    sums[kIdx * NSLOT + j] = tot;
    s[j] = tot;
  }
  __syncthreads();
  if (j == 0 && kIdx < 20) {
    float As[16], Bs[4];
    for (int i = 0; i < 4; ++i)
      for (int jj = 0; jj < 4; ++jj) As[4 * i + jj] = 0.5f * (invK[4 * i + jj] + invK[4 * jj + i]);
    for (int i = 0; i < 2; ++i)
      for (int jj = 0; jj < 2; ++jj) Bs[2 * i + jj] = 0.5f * (invKb[2 * i + jj] + invKb[2 * jj + i]);
    const float* th = thetaAll + kIdx * 12;
    float g[12];
    float v2v[4] = {th[6] - 1.f, th[7], th[8], th[9] - 1.f};
    for (int i = 0; i < 4; ++i) {
      float a1 = 0.f, a2 = 0.f;
      for (int jj = 0; jj < 4; ++jj) { a1 += As[4 * i + jj] * th[jj]; a2 += As[4 * i + jj] * v2v[jj]; }
      g[i] = a1 - c * s[i];            // dL/dT1
      g[6 + i] = a2 - c * s[4 + i];    // dL/dT2
    }
    for (int i = 0; i < 2; ++i) {
      float g1 = 0.f, g2 = 0.f;
      for (int jj = 0; jj < 2; ++jj) { g1 += Bs[2 * i + jj] * th[4 + jj]; g2 += Bs[2 * i + jj] * th[10 + jj]; }
      g[4 + i] = g1 - c * s[10 + i];   // dL/db1
      g[10 + i] = g2 - c * s[8 + i];   // dL/db2
    }
    float* thn = thetaAll + (kIdx + 1) * 12;
    for (int i = 0; i < 12; ++i) thn[i] = th[i] - LR_CONST * g[i];
  }
}

// --- backward adjoint recursion: lambda_k = (I - LR*H_k) lambda_{k+1} -------
__global__ void backward_kernel(const float* __restrict__ sums,
                                const float* __restrict__ thetaAll,
                                float* __restrict__ lambdaAll,
                                const float* __restrict__ invK,
                                const float* __restrict__ invKb,
                                float c) {
  if (threadIdx.x != 0 || blockIdx.x != 0) return;
  float As[16], Bs[4];
  for (int i = 0; i < 4; ++i)
    for (int j = 0; j < 4; ++j) As[4 * i + j] = 0.5f * (invK[4 * i + j] + invK[4 * j + i]);
  for (int i = 0; i < 2; ++i)
    for (int j = 0; j < 2; ++j) Bs[2 * i + j] = 0.5f * (invKb[2 * i + j] + invKb[2 * j + i]);

  float lam[12];
  {  // lambda_20 = grad L at Theta_20
    const float* th = thetaAll + 20 * 12;
    const float* s = sums + 20 * NSLOT;
    float v2v[4] = {th[6] - 1.f, th[7], th[8], th[9] - 1.f};
    for (int i = 0; i < 4; ++i) {
      float a1 = 0.f, a2 = 0.f;
      for (int j = 0; j < 4; ++j) { a1 += As[4 * i + j] * th[j]; a2 += As[4 * i + j] * v2v[j]; }
      lam[i] = a1 - c * s[i];
      lam[6 + i] = a2 - c * s[4 + i];
    }
    for (int i = 0; i < 2; ++i) {
      float g1 = 0.f, g2 = 0.f;
      for (int j = 0; j < 2; ++j) { g1 += Bs[2 * i + j] * th[4 + j]; g2 += Bs[2 * i + j] * th[10 + j]; }
      lam[4 + i] = g1 - c * s[10 + i];
      lam[10 + i] = g2 - c * s[8 + i];
    }
    for (int i = 0; i < 12; ++i) lambdaAll[20 * 12 + i] = lam[i];
  }
  for (int k = 19; k >= 0; --k) {
    const float* sk = sums + k * NSLOT;
    float H[12][12];
    for (int i = 0; i < 12; ++i)
      for (int j = 0; j < 12; ++j) H[i][j] = 0.f;
    for (int i = 0; i < 4; ++i)
      for (int j = 0; j < 4; ++j) { H[i][j] = As[4 * i + j]; H[6 + i][6 + j] = As[4 * i + j]; }
    for (int i = 0; i < 2; ++i)
      for (int j = 0; j < 2; ++j) { H[4 + i][4 + j] = Bs[2 * i + j]; H[10 + i][10 + j] = Bs[2 * i + j]; }
    for (int a = 0; a < 2; ++a)
      for (int b = 0; b < 2; ++b)
        for (int e = 0; e < 2; ++e) {  // T1(a,b) x T2(b,e)
          float v = c * sk[12 + 4 * a + 2 * b + e];
          H[2 * a + b][6 + 2 * b + e] -= v; H[6 + 2 * b + e][2 * a + b] -= v;
        }
    for (int a = 0; a < 2; ++a)
      for (int b = 0; b < 2; ++b) {      // T1(a,b) x b2(b)
        float v = c * sk[20 + 2 * a + b];
        H[2 * a + b][10 + b] -= v; H[10 + b][2 * a + b] -= v;
      }
    for (int a = 0; a < 2; ++a)
      for (int b = 0; b < 2; ++b)
        for (int e = 0; e < 2; ++e)      // T2(a,b) x T2(a,e): full block
          H[6 + 2 * a + b][6 + 2 * a + e] -= c * sk[24 + 4 * a + 2 * b + e];
    for (int a = 0; a < 2; ++a)
      for (int b = 0; b < 2; ++b) {      // T2(a,b) x b2(a)
        float v = c * sk[32 + 2 * a + b];
        H[6 + 2 * a + b][10 + a] -= v; H[10 + a][6 + 2 * a + b] -= v;
      }
    for (int a = 0; a < 2; ++a) H[10 + a][10 + a] -= c * sk[36 + a];
    float nl2[12];
    for (int i = 0; i < 12; ++i) {
      float hv = 0.f;
      for (int j = 0; j < 12; ++j) hv += H[i][j] * lam[j];
      nl2[i] = lam[i] - LR_CONST * hv;
    }
    for (int i = 0; i < 12; ++i) { lam[i] = nl2[i]; lambdaAll[k * 12 + i] = lam[i]; }
  }
}

// --- per-sample outputs: dot_q, dot_p, dot_x --------------------------------
__global__ void output_kernel(const float* __restrict__ inp, int K,
                              const float* __restrict__ thetaAll,
                              const float* __restrict__ lambdaAll,
                              float c, float* __restrict__ out) {
  __shared__ float thS[21 * 12];
  __shared__ float laS[21 * 12];
  for (int j = threadIdx.x; j < 252; j += blockDim.x) { thS[j] = thetaAll[j]; laS[j] = lambdaAll[j]; }
  __syncthreads();
  const float2* in2 = (const float2*)inp;
  float2* out2 = (float2*)out;
  const float LRc = LR_CONST * c;
  const float* thN = thS + 20 * 12;
  const float T10 = thN[0], T11 = thN[1], T12 = thN[2], T13 = thN[3];
  const float b10 = thN[4], b11 = thN[5];
  const float T20 = thN[6], T21 = thN[7], T22 = thN[8], T23 = thN[9];
  const float b20 = thN[10], b21 = thN[11];
  const int stride = gridDim.x * blockDim.x;
  for (int i = blockIdx.x * blockDim.x + threadIdx.x; i < K; i += stride) {
    float2 qv = in2[i], pv = in2[K + i], xv = in2[2 * K + i];
    float q0 = qv.x, q1 = qv.y, p0 = pv.x, p1 = pv.y, x0 = xv.x, x1 = xv.y;
    // dot_q = T1 tanh(T2 q + b2) + b1 at Theta_20
    float u0 = fmaf(T20, q0, fmaf(T21, q1, b20));
    float u1 = fmaf(T22, q0, fmaf(T23, q1, b21));
    float h0 = tanhf(u0), h1 = tanhf(u1);
    float dq0 = fmaf(T10, h0, fmaf(T11, h1, b10));
    float dq1 = fmaf(T12, h0, fmaf(T13, h1, b11));
    // dot_x
    float ux0 = fmaf(T20, x0, fmaf(T21, x1, b20));
    float ux1 = fmaf(T22, x0, fmaf(T23, x1, b21));
    float hx0 = tanhf(ux0), hx1 = tanhf(ux1);
    float dx0 = fmaf(T10, hx0, fmaf(T11, hx1, b10));
    float dx1 = fmaf(T12, hx0, fmaf(T13, hx1, b11));
    // dot_p direct term: -c * T2N^T rN
    float s0 = 1.f - h0 * h0, s1 = 1.f - h1 * h1;
    float w0 = fmaf(T10, p0, T12 * p1);
    float w1 = fmaf(T11, p0, T13 * p1);
    float r0 = s0 * w0, r1 = s1 * w1;
    float gp0 = -c * (T20 * r0 + T22 * r1);
    float gp1 = -c * (T21 * r0 + T23 * r1);
    // + LR*c * sum_k [ T2_k^T(s.(L1^T p) + t.(L2 q + lb2)) + L2^T r ]
    for (int k = 0; k < 20; ++k) {
      const float* th = thS + k * 12;
      const float* lm = laS + (k + 1) * 12;
      float uu0 = fmaf(th[6], q0, fmaf(th[7], q1, th[10]));
      float uu1 = fmaf(th[8], q0, fmaf(th[9], q1, th[11]));
      float hh0 = tanhf(uu0), hh1 = tanhf(uu1);
      float ss0 = 1.f - hh0 * hh0, ss1 = 1.f - hh1 * hh1;
      float ww0 = fmaf(th[0], p0, th[2] * p1);
      float ww1 = fmaf(th[1], p0, th[3] * p1);
      float rr0 = ss0 * ww0, rr1 = ss1 * ww1;
      float tt0 = -2.f * hh0 * rr0, tt1 = -2.f * hh1 * rr1;
      float a0 = ss0 * fmaf(lm[0], p0, lm[2] * p1);
      float a1 = ss1 * fmaf(lm[1], p0, lm[3] * p1);
      float e0 = tt0 * fmaf(lm[6], q0, fmaf(lm[7], q1, lm[10]));
      float e1 = tt1 * fmaf(lm[8], q0, fmaf(lm[9], q1, lm[11]));
      float f0 = a0 + e0, f1 = a1 + e1;
      gp0 += LRc * (th[6] * f0 + th[8] * f1 + lm[6] * rr0 + lm[8] * rr1);
      gp1 += LRc * (th[7] * f0 + th[9] * f1 + lm[7] * rr0 + lm[9] * rr1);
    }
    float2 o;
    o.x = dq0; o.y = dq1; out2[i] = o;
    o.x = gp0; o.y = gp1; out2[K + i] = o;
    o.x = dx0; o.y = dx1; out2[2 * K + i] = o;
  }
}

extern "C" void kernel_launch(void* const* d_in, const int* in_sizes, int n_in,
                              void* d_out, int out_size, void* d_ws, size_t ws_size,
                              hipStream_t stream) {
  (void)n_in; (void)out_size;
  const float* inp = (const float*)d_in[1];
  const float* t1i = (const float*)d_in[2];
  const float* b1i = (const float*)d_in[3];
  const float* t2i = (const float*)d_in[4];
  const float* b2i = (const float*)d_in[5];
  const float* invK = (const float*)d_in[6];
  const float* invKb = (const float*)d_in[7];
  const int K = in_sizes[1] / 6;
  const float c = 1.0f / (2.0f * (float)K);

  float* wsf = (float*)d_ws;
  float* thetaAll = wsf;              // 21*12
  float* lambdaAll = wsf + 256;       // 21*12
  float* sums = wsf + 512;            // 21*NSLOT
  float* partials = wsf + 2048;       // NB*NSLOT

  int NB = 512;
  size_t avail = ws_size / 4;
  if (avail < 2048 + (size_t)NB * NSLOT) {
    long room = (long)avail - 2048;
    NB = room > NSLOT ? (int)(room / NSLOT) : 1;
  }

  init_theta_kernel<<<1, 32, 0, stream>>>(t1i, b1i, t2i, b2i, thetaAll);
  for (int k = 0; k <= 20; ++k) {
    pass_reduce_kernel<<<NB, 256, 0, stream>>>(inp, K, thetaAll, k, partials);
    reduce_update_kernel<<<1, 64, 0, stream>>>(partials, NB, sums, k, thetaAll, invK, invKb, c);
  }
  backward_kernel<<<1, 32, 0, stream>>>(sums, thetaAll, lambdaAll, invK, invKb, c);
  int OB = (K + 255) / 256;
  if (OB > 4096) OB = 4096;
  if (OB < 1) OB = 1;
  output_kernel<<<OB, 256, 0, stream>>>(inp, K, thetaAll, lambdaAll, c, (float*)d_out);
}